// RNA_embedding_33414845562889
// MI455X (gfx1250) — compile-verified
//
#include <hip/hip_runtime.h>

// Problem constants (from the reference)
#define VOCAB 8
#define EMB   128
#define BATCH 4
#define SEQ   512

typedef __attribute__((ext_vector_type(2))) float v2f;
typedef __attribute__((ext_vector_type(4))) float v4f;
typedef __attribute__((ext_vector_type(8))) float v8f;

// ---------------------------------------------------------------------------
// Kernel 1: P1 = table @ W[:128], P2b = table @ W[128:] + bias
// via V_WMMA_F32_16X16X4_F32. table is padded M=8..15 with zeros, done
// branch-free (masked multiply) so EXEC stays all-1s — no saveexec churn
// around the WMMAs.
//
// ISA VGPR layouts (CDNA5 ISA §7.12.2):
//   A 16x4 (MxK): lanes 0-15 -> M=lane, VGPR0 K=0, VGPR1 K=1;
//                 lanes 16-31 -> M=lane-16, VGPR0 K=2, VGPR1 K=3.
//   B 4x16 (KxN): mirrored (lanes 0-15 -> N=lane, VGPR0 K=0, VGPR1 K=1;
//                 lanes 16-31 -> VGPR0 K=2, VGPR1 K=3).
//   C/D 16x16:    VGPR r: lanes 0-15 -> M=r, lanes 16-31 -> M=r+8.
// ---------------------------------------------------------------------------
__global__ __launch_bounds__(512)
void RNA_emb_precompute_wmma(const float* __restrict__ table,
                             const float* __restrict__ W,
                             const float* __restrict__ bias,
                             float* __restrict__ P)   // [2][VOCAB][EMB]
{
    const int tid  = threadIdx.x;
    const int wave = tid >> 5;
    const int lane = tid & 31;
    const int m    = wave >> 3;            // 0 -> W1, 1 -> W2
    const int nt   = wave & 7;             // 16-wide column tile of EMB
    const int mrow = lane & 15;            // A-matrix row this lane holds
    const int kofs = (lane >> 4) << 1;     // 0 for lanes 0-15, 2 for 16-31
    const int ncol = nt * 16 + (lane & 15);

    // Branch-free zero padding for A rows 8..15: clamp the address into the
    // valid table, kill the value with a 0/1 mask (v_cndmask + v_mul, no EXEC).
    const float amask = (mrow < VOCAB) ? 1.0f : 0.0f;
    const float* Arow = table + (mrow & (VOCAB - 1)) * EMB;
    const float* Wm   = W + m * EMB * EMB;  // W is (2*EMB, EMB) row-major

    v8f acc = {};
    #pragma unroll 4
    for (int k0 = 0; k0 < EMB; k0 += 4) {
        // A fragment: two consecutive K elements -> one 8-B load, then mask.
        v2f a = *(const v2f*)&Arow[k0 + kofs];
        a *= amask;
        // B fragment: two rows of Wm, same column.
        v2f b;
        b.x = Wm[(k0 + kofs)     * EMB + ncol];
        b.y = Wm[(k0 + kofs + 1) * EMB + ncol];
        // 8 args: (neg_a, A, neg_b, B, c_mod, C, reuse_a, reuse_b)
        acc = __builtin_amdgcn_wmma_f32_16x16x4_f32(
                  false, a, false, b, (short)0, acc, false, false);
    }

    // Rows M=0..7 of the result live in lanes 0-15, acc components 0..7.
    if (lane < 16) {
        const float bb = (m == 1) ? bias[ncol] : 0.0f;
        #pragma unroll
        for (int r = 0; r < VOCAB; ++r) {
            P[m * (VOCAB * EMB) + r * EMB + ncol] = acc[r] + bb;
        }
    }
}

// ---------------------------------------------------------------------------
// Kernel 2: out[b,i,j,:] = P1[x[b,i]] + P2b[x[b,j]]
// One block per (b,i). Build the 8x128 slice S[c][:] = P1[a] + P2b[c] in LDS
// (a = x[b,i] fixed per block), cache x[b,:] in LDS, then stream 512 rows of
// 512 B each as coalesced per-lane 16-B non-temporal stores.
// Pure store-bandwidth kernel: 512 MiB writes, ~10 KB reads per block.
// ---------------------------------------------------------------------------
__global__ __launch_bounds__(256)
void RNA_emb_broadcast_store(const int* __restrict__ x,
                             const float* __restrict__ P,   // [2][VOCAB][EMB]
                             float* __restrict__ out)       // [B][SEQ][SEQ][EMB]
{
    __shared__ __align__(16) float S[VOCAB * EMB];   // 4 KB
    __shared__ int cidx[SEQ];                        // 2 KB

    const int bi  = blockIdx.x;        // b*SEQ + i
    const int tid = threadIdx.x;
    const int b   = bi >> 9;           // SEQ == 512

    const int a = x[bi];               // row index for the i side (uniform)

    // S[c*EMB + n] = P1[a][n] + (P2[c][n] + bias[n])
    for (int t = tid; t < VOCAB * EMB; t += 256) {
        const int n = t & (EMB - 1);
        S[t] = P[a * EMB + n] + P[VOCAB * EMB + t];
    }
    for (int t = tid; t < SEQ; t += 256) {
        cidx[t] = x[b * SEQ + t];
    }
    __syncthreads();

    const int wave = tid >> 5;
    const int lane = tid & 31;
    float* rowbase = out + (size_t)bi * (SEQ * EMB);

    // Each wave writes one 128-float row per iteration: lane -> 16 B slice.
    #pragma unroll 4
    for (int j = wave; j < SEQ; j += 8) {
        const int c = cidx[j];
        const v4f v = *(const v4f*)&S[c * EMB + lane * 4];
        v4f* dst = (v4f*)(rowbase + (size_t)j * EMB + lane * 4);
        __builtin_nontemporal_store(v, dst);   // NT: don't thrash 192MB L2
    }
}

// ---------------------------------------------------------------------------
// Launch: inputs in setup_inputs() order: x(int32), table(f32), W(f32), b(f32)
// ---------------------------------------------------------------------------
extern "C" void kernel_launch(void* const* d_in, const int* in_sizes, int n_in,
                              void* d_out, int out_size, void* d_ws, size_t ws_size,
                              hipStream_t stream) {
    const int*   x     = (const int*)d_in[0];
    const float* table = (const float*)d_in[1];
    const float* W     = (const float*)d_in[2];
    const float* bias  = (const float*)d_in[3];
    float*       out   = (float*)d_out;
    float*       P     = (float*)d_ws;   // 2*VOCAB*EMB floats = 8 KB scratch

    hipLaunchKernelGGL(RNA_emb_precompute_wmma, dim3(1), dim3(512), 0, stream,
                       table, W, bias, P);
    hipLaunchKernelGGL(RNA_emb_broadcast_store, dim3(BATCH * SEQ), dim3(256), 0, stream,
                       x, P, out);
}